// GIN_13005160973224
// MI455X (gfx1250) — compile-verified
//
#include <hip/hip_runtime.h>

#define N_NODES 100000
#define N_EDGES 1600000
#define FEAT    128
#define LDS_STR 132   // 128 + 4 pad: A-frag LDS reads hit distinct banks
#define MROWS   32    // rows per workgroup: 2 M-tiles per wave, B-frag reuse x2

typedef __attribute__((ext_vector_type(2))) float v2f;
typedef __attribute__((ext_vector_type(8))) float v8f;

// ---------------------------------------------------------------------------
// tmp = in   (vectorized d2d copy; seeds agg with the self term x_i)
// ---------------------------------------------------------------------------
__global__ __launch_bounds__(256) void gin_copy_f4(const float4* __restrict__ in,
                                                   float4* __restrict__ out, int n4) {
    int i = blockIdx.x * 256 + threadIdx.x;
    if (i < n4) out[i] = in[i];
}

// ---------------------------------------------------------------------------
// acc[dst[e], :] += H[src[e], :]   one wave32 per edge, float4 per lane,
// 4x global_atomic_add_f32 per lane (resolved in L2; H is L2-resident).
// ---------------------------------------------------------------------------
__global__ __launch_bounds__(256) void gin_scatter_add(const float* __restrict__ H,
                                                       const long long* __restrict__ src,
                                                       const long long* __restrict__ dst,
                                                       float* __restrict__ acc) {
    int e = (blockIdx.x * 256 + threadIdx.x) >> 5;
    if (e >= N_EDGES) return;
    int lane = threadIdx.x & 31;
    int s = (int)src[e];
    int d = (int)dst[e];
    const float4 v = *(const float4*)(H + (size_t)s * FEAT + lane * 4);
    float* p = acc + (size_t)d * FEAT + lane * 4;
    unsafeAtomicAdd(p + 0, v.x);
    unsafeAtomicAdd(p + 1, v.y);
    unsafeAtomicAdd(p + 2, v.z);
    unsafeAtomicAdd(p + 3, v.w);
}

// ---------------------------------------------------------------------------
// WMMA f32 16x16x4 dual-M-tile GEMM core (wave32, §7.12.2 layouts):
//   A 16x4 : lane m = lane&15, VGPR j holds K = 4s + 2*(lane>>4) + j
//   B 4x16 : lane n = lane&15, VGPR j holds K = 4s + 2*(lane>>4) + j
//   C 16x16: lane n = lane&15, VGPR i holds M = i + 8*(lane>>4)
// One B fragment feeds both row-tiles (rows 0..15 and 16..31).
// ---------------------------------------------------------------------------
__device__ __forceinline__ void mm128_dual(const float* __restrict__ lds_a,
                                           const float* __restrict__ W,
                                           int n0, int ln, int hi, float bias,
                                           v8f& c0, v8f& c1) {
#pragma unroll
    for (int i = 0; i < 8; ++i) { c0[i] = bias; c1[i] = bias; }  // bias is per-column
    const int n = n0 + ln;
    const float* a_lo = lds_a + ln * LDS_STR;
    const float* a_hi = lds_a + (ln + 16) * LDS_STR;
#pragma unroll 8
    for (int s = 0; s < 32; ++s) {
        const int kb = 4 * s + 2 * hi;
        v2f b, a0, a1;
        b.x  = W[(size_t)kb * FEAT + n];
        b.y  = W[(size_t)(kb + 1) * FEAT + n];
        a0.x = a_lo[kb];
        a0.y = a_lo[kb + 1];
        a1.x = a_hi[kb];
        a1.y = a_hi[kb + 1];
        c0 = __builtin_amdgcn_wmma_f32_16x16x4_f32(false, a0, false, b,
                                                   (short)0, c0, false, false);
        c1 = __builtin_amdgcn_wmma_f32_16x16x4_f32(false, a1, false, b,
                                                   (short)0, c1, false, false);
    }
}

// cooperative 32x128 tile load: thread t -> row t>>3, cols (t&7)*16 .. +15
__device__ __forceinline__ void load_tile32(const float* __restrict__ X,
                                            float* __restrict__ Xt,
                                            int row0, int t) {
    const int r = t >> 3, c = (t & 7) * 16;
    const float4* g = (const float4*)(X + (size_t)(row0 + r) * FEAT + c);
    float4* l = (float4*)(Xt + r * LDS_STR + c);
#pragma unroll
    for (int q = 0; q < 4; ++q) l[q] = g[q];
}

// ---------------------------------------------------------------------------
// Y = relu(X@W1 + b1) @ W2 + b2     X,Y: [N,128]; 32 rows per block, 8 waves,
// wave w owns N-tile [16w,16w+16) in both GEMMs; mid staged in LDS.
// ---------------------------------------------------------------------------
__global__ __launch_bounds__(256) void gin_mlp128(const float* __restrict__ X,
                                                  const float* __restrict__ W1,
                                                  const float* __restrict__ B1,
                                                  const float* __restrict__ W2,
                                                  const float* __restrict__ B2,
                                                  float* __restrict__ Y) {
    __shared__ float Xt[MROWS * LDS_STR];
    __shared__ float Mid[MROWS * LDS_STR];
    const int row0 = blockIdx.x * MROWS;
    const int t = threadIdx.x;

    load_tile32(X, Xt, row0, t);
    __syncthreads();

    const int wave = t >> 5, lane = t & 31;
    const int n0 = wave * 16, ln = lane & 15, hi = lane >> 4;

    // GEMM 1: relu -> Mid
    {
        v8f c0, c1;
        mm128_dual(Xt, W1, n0, ln, hi, B1[n0 + ln], c0, c1);
#pragma unroll
        for (int i = 0; i < 8; ++i) {
            const int m = i + 8 * hi;
            const float v0 = c0[i], v1 = c1[i];
            Mid[m * LDS_STR + n0 + ln]        = v0 > 0.0f ? v0 : 0.0f;
            Mid[(m + 16) * LDS_STR + n0 + ln] = v1 > 0.0f ? v1 : 0.0f;
        }
    }
    __syncthreads();

    // GEMM 2 -> global
    {
        v8f d0, d1;
        mm128_dual(Mid, W2, n0, ln, hi, B2[n0 + ln], d0, d1);
#pragma unroll
        for (int i = 0; i < 8; ++i) {
            const int m = i + 8 * hi;
            Y[(size_t)(row0 + m) * FEAT + n0 + ln]      = d0[i];
            Y[(size_t)(row0 + m + 16) * FEAT + n0 + ln] = d1[i];
        }
    }
}

// ---------------------------------------------------------------------------
// Final layer: y = relu(X@W1 + b1) @ w2 + b2   with w2: [128,1] -> y: [N]
// GEMM1 via WMMA into LDS, then a 128-wide dot per row.
// ---------------------------------------------------------------------------
__global__ __launch_bounds__(256) void gin_mlp_final(const float* __restrict__ X,
                                                     const float* __restrict__ W1,
                                                     const float* __restrict__ B1,
                                                     const float* __restrict__ w2,
                                                     const float* __restrict__ b2,
                                                     float* __restrict__ Y) {
    __shared__ float Xt[MROWS * LDS_STR];
    __shared__ float Mid[MROWS * LDS_STR];
    __shared__ float red[MROWS][9];
    const int row0 = blockIdx.x * MROWS;
    const int t = threadIdx.x;

    load_tile32(X, Xt, row0, t);
    __syncthreads();

    const int wave = t >> 5, lane = t & 31;
    const int n0 = wave * 16, ln = lane & 15, hi = lane >> 4;

    {
        v8f c0, c1;
        mm128_dual(Xt, W1, n0, ln, hi, B1[n0 + ln], c0, c1);
#pragma unroll
        for (int i = 0; i < 8; ++i) {
            const int m = i + 8 * hi;
            const float v0 = c0[i], v1 = c1[i];
            Mid[m * LDS_STR + n0 + ln]        = v0 > 0.0f ? v0 : 0.0f;
            Mid[(m + 16) * LDS_STR + n0 + ln] = v1 > 0.0f ? v1 : 0.0f;
        }
    }
    __syncthreads();

    // row dot: thread t covers row m = t>>3, k-chunk (t&7)*16 .. +15
    {
        const int m = t >> 3, seg = t & 7, k0 = seg * 16;
        float s = 0.0f;
#pragma unroll
        for (int j = 0; j < 16; ++j) s += Mid[m * LDS_STR + k0 + j] * w2[k0 + j];
        red[m][seg] = s;
    }
    __syncthreads();
    if (t < MROWS) {
        float s = b2[0];
#pragma unroll
        for (int j = 0; j < 8; ++j) s += red[t][j];
        Y[row0 + t] = s;
    }
}

// ---------------------------------------------------------------------------
// Inputs: x, edge_index, w11,b11,w12,b12, w21,b21,w22,b22, w31,b31,w32,b32
// ---------------------------------------------------------------------------
extern "C" void kernel_launch(void* const* d_in, const int* in_sizes, int n_in,
                              void* d_out, int out_size, void* d_ws, size_t ws_size,
                              hipStream_t stream) {
    const float* x = (const float*)d_in[0];
    const long long* ei = (const long long*)d_in[1];   // jnp.int64 in reference
    const long long* src = ei;
    const long long* dst = ei + N_EDGES;
    const float* w11 = (const float*)d_in[2];  const float* b11 = (const float*)d_in[3];
    const float* w12 = (const float*)d_in[4];  const float* b12 = (const float*)d_in[5];
    const float* w21 = (const float*)d_in[6];  const float* b21 = (const float*)d_in[7];
    const float* w22 = (const float*)d_in[8];  const float* b22 = (const float*)d_in[9];
    const float* w31 = (const float*)d_in[10]; const float* b31 = (const float*)d_in[11];
    const float* w32 = (const float*)d_in[12]; const float* b32 = (const float*)d_in[13];
    float* out = (float*)d_out;

    float* tmp = (float*)d_ws;                         // [N_NODES,128] agg buffer
    float* h   = tmp + (size_t)N_NODES * FEAT;         // [N_NODES,128] hidden

    const int n4 = N_NODES * FEAT / 4;
    const int copy_blocks    = (n4 + 255) / 256;
    const int scatter_blocks = (N_EDGES + 7) / 8;      // 8 edges (waves) / block
    const int mlp_blocks     = N_NODES / MROWS;        // 100000 % 32 == 0

    // ---- layer 1 ----
    gin_copy_f4<<<copy_blocks, 256, 0, stream>>>((const float4*)x, (float4*)tmp, n4);
    gin_scatter_add<<<scatter_blocks, 256, 0, stream>>>(x, src, dst, tmp);
    gin_mlp128<<<mlp_blocks, 256, 0, stream>>>(tmp, w11, b11, w12, b12, h);
    // ---- layer 2 ----
    gin_copy_f4<<<copy_blocks, 256, 0, stream>>>((const float4*)h, (float4*)tmp, n4);
    gin_scatter_add<<<scatter_blocks, 256, 0, stream>>>(h, src, dst, tmp);
    gin_mlp128<<<mlp_blocks, 256, 0, stream>>>(tmp, w21, b21, w22, b22, h);
    // ---- layer 3 ----
    gin_copy_f4<<<copy_blocks, 256, 0, stream>>>((const float4*)h, (float4*)tmp, n4);
    gin_scatter_add<<<scatter_blocks, 256, 0, stream>>>(h, src, dst, tmp);
    gin_mlp_final<<<mlp_blocks, 256, 0, stream>>>(tmp, w31, b31, w32, b32, out);
}